// HybridMixSTEDecoder_69887707841118
// MI455X (gfx1250) — compile-verified
//
#include <hip/hip_runtime.h>

typedef __attribute__((ext_vector_type(2))) float v2f;
typedef __attribute__((ext_vector_type(8))) float v8f;

namespace {

constexpr int kHid = 512;   // hidden dim (K)
constexpr int kG   = 5;     // groups
constexpr int kTp  = 27;    // T_patches
constexpr int kPt  = 9;     // patch_size
constexpr int kC   = 3;     // out channels
constexpr int kT   = 243;   // kTp * kPt
constexpr int kJ   = 17;    // joints
constexpr int kWavesPerBlock = 8;
constexpr int kRows = 1024 * kTp;  // 27648 GEMM rows (B * T_patches)

struct Joints { int j[5]; };

// One wave owns a 16-row M-tile and all NT column tiles of its group.
// D = A(16x512) x B(512x16*NT) via chained V_WMMA_F32_16X16X4_F32.
template <int GS, int NT>
__global__ __launch_bounds__(256)
void group_head_gemm(const float* __restrict__ tokens,
                     const float* __restrict__ W,
                     const float* __restrict__ bias,
                     float* __restrict__ out,
                     int g, Joints joints) {
  constexpr int OUT_DIM = kPt * GS * kC;   // 81 or 135

  const int wave = threadIdx.x >> 5;
  const int lane = threadIdx.x & 31;
  const int m0   = (blockIdx.x * kWavesPerBlock + wave) * 16;  // global row base
  const int idx  = lane & 15;          // row (A) / col (B) within tile
  const int koff = (lane >> 4) * 2;    // K sub-offset per ISA 16x4 A layout

  // A: tokens[(m0+idx), g, koff + k0 ...], row stride G*HID
  const float* Arow =
      tokens + (size_t)(m0 + idx) * (kG * kHid) + (size_t)g * kHid + koff;

  // B: W[n, koff + k0 ...], 32-bit offsets from scalar base (OOB cols clamped)
  int boff[NT];
#pragma unroll
  for (int nt = 0; nt < NT; ++nt) {
    int n = nt * 16 + idx;
    if (n >= OUT_DIM) n = OUT_DIM - 1;
    boff[nt] = n * kHid + koff;
  }

  v8f acc[NT];
  const v8f zero = {0.f, 0.f, 0.f, 0.f, 0.f, 0.f, 0.f, 0.f};
#pragma unroll
  for (int nt = 0; nt < NT; ++nt) acc[nt] = zero;

#pragma unroll 2
  for (int k0 = 0; k0 < kHid; k0 += 4) {
    // tokens are touched exactly once -> non-temporal
    v2f a = __builtin_nontemporal_load((const v2f*)(Arow + k0));
#pragma unroll
    for (int nt = 0; nt < NT; ++nt) {
      v2f b = *(const v2f*)(W + boff[nt] + k0);
      acc[nt] = __builtin_amdgcn_wmma_f32_16x16x4_f32(
          false, a, false, b, (short)0, acc[nt], false, false);
    }
  }

  // Epilogue: C/D layout -> row = r + (lane>=16 ? 8 : 0), col = lane&15.
  // Scatter col o -> (pt, p, c), joint = group[p]; groups are disjoint so
  // every output element is written exactly once (count == 1 in reference).
  const int rbase = (lane >> 4) << 3;
#pragma unroll
  for (int nt = 0; nt < NT; ++nt) {
    const int o = nt * 16 + idx;
    if (o < OUT_DIM) {
      const float bv  = bias[o];
      const int pt    = o / (GS * kC);
      const int rem   = o - pt * (GS * kC);
      const int p     = rem / kC;
      const int c     = rem - p * kC;
      const int joint = joints.j[p];
#pragma unroll
      for (int r = 0; r < 8; ++r) {
        const int m  = m0 + rbase + r;
        const int bb = m / kTp;
        const int tp = m - bb * kTp;
        const int t  = tp * kPt + pt;
        const int oidx = ((bb * kT + t) * kJ + joint) * kC + c;
        __builtin_nontemporal_store(acc[nt][r] + bv, out + oidx);
      }
    }
  }
}

}  // namespace

extern "C" void kernel_launch(void* const* d_in, const int* in_sizes, int n_in,
                              void* d_out, int out_size, void* d_ws, size_t ws_size,
                              hipStream_t stream) {
  const float* tokens = (const float*)d_in[0];
  float* out = (float*)d_out;

  const dim3 block(256);
  const dim3 grid(kRows / (kWavesPerBlock * 16));  // 216 blocks per group

  const Joints j0 = {{1, 2, 3, 0, 0}};
  const Joints j1 = {{4, 5, 6, 0, 0}};
  const Joints j2 = {{0, 7, 8, 9, 10}};
  const Joints j3 = {{11, 12, 13, 0, 0}};
  const Joints j4 = {{14, 15, 16, 0, 0}};

  group_head_gemm<3, 6><<<grid, block, 0, stream>>>(
      tokens, (const float*)d_in[1], (const float*)d_in[2], out, 0, j0);
  group_head_gemm<3, 6><<<grid, block, 0, stream>>>(
      tokens, (const float*)d_in[3], (const float*)d_in[4], out, 1, j1);
  group_head_gemm<5, 9><<<grid, block, 0, stream>>>(
      tokens, (const float*)d_in[5], (const float*)d_in[6], out, 2, j2);
  group_head_gemm<3, 6><<<grid, block, 0, stream>>>(
      tokens, (const float*)d_in[7], (const float*)d_in[8], out, 3, j3);
  group_head_gemm<3, 6><<<grid, block, 0, stream>>>(
      tokens, (const float*)d_in[9], (const float*)d_in[10], out, 4, j4);
}